// BinaryTokenClassificationModel_66881230733873
// MI455X (gfx1250) — compile-verified
//
#include <hip/hip_runtime.h>
#include <hip/hip_bf16.h>

// Problem constants (from reference): B=16, L=1024, H=1024, V=50257, S=512
#define Bsz 16
#define Lsz 1024
#define Hsz 1024
#define Ssz 512

typedef __attribute__((ext_vector_type(2))) float v2f;
typedef __attribute__((ext_vector_type(8))) float v8f;

// ---------------------------------------------------------------------------
// Kernel 0: zero the scalar accumulators in workspace (graph-capture safe).
// ---------------------------------------------------------------------------
__global__ __launch_bounds__(256) void zero_ws_kernel(float* __restrict__ ws, int n) {
    int i = blockIdx.x * 256 + threadIdx.x;
    if (i < n) ws[i] = 0.0f;
}

// ---------------------------------------------------------------------------
// Kernel 1: per-token dot products + scalar segment-sum by word id.
// One wave (32 lanes) per token. Each lane reads 8 float4 = 32 floats of the
// embedding row, dots against w_src / w_tgt chunks, wave32 shuffle-reduce,
// lane 0 does 4 scalar atomicAdds. This is the bandwidth-dominant kernel
// (64 MB of embedding gathers) -> plain vectorized VMEM is optimal here.
// ---------------------------------------------------------------------------
__global__ __launch_bounds__(256) void token_dot_kernel(
    const int* __restrict__ ids, const int* __restrict__ mask,
    const long long* __restrict__ swid, const long long* __restrict__ twid,
    const float* __restrict__ embed, const float* __restrict__ w,
    float* __restrict__ a_sum, float* __restrict__ a_cnt,
    float* __restrict__ c_sum, float* __restrict__ c_cnt)
{
    const int lane = threadIdx.x & 31;
    const int wave = threadIdx.x >> 5;
    const int tok  = blockIdx.x * 8 + wave;          // 0 .. B*L-1
    const int b    = tok >> 10;                      // tok / L

    const long long row_id = (long long)ids[tok];
    const float4* __restrict__ row = (const float4*)(embed + (size_t)row_id * Hsz);
    const float4* __restrict__ ws4 = (const float4*)(w);
    const float4* __restrict__ wt4 = (const float4*)(w + Hsz);

    float ps = 0.0f, pt = 0.0f;
#pragma unroll
    for (int i = 0; i < 8; ++i) {
        const int idx = lane + i * 32;               // covers 256 float4 = 1024 floats
        float4 e  = row[idx];
        float4 s4 = ws4[idx];
        float4 t4 = wt4[idx];
        ps += e.x * s4.x + e.y * s4.y + e.z * s4.z + e.w * s4.w;
        pt += e.x * t4.x + e.y * t4.y + e.z * t4.z + e.w * t4.w;
    }
#pragma unroll
    for (int off = 16; off > 0; off >>= 1) {
        ps += __shfl_down(ps, off, 32);
        pt += __shfl_down(pt, off, 32);
    }
    if (lane == 0 && mask[tok] > 0) {
        const int sw = (int)swid[tok];
        const int tw = (int)twid[tok];
        if (sw >= 0 && sw < Ssz) {
            atomicAdd(&a_sum[b * Ssz + sw], ps);
            atomicAdd(&a_cnt[b * Ssz + sw], 1.0f);
        }
        if (tw >= 0 && tw < Ssz) {
            atomicAdd(&c_sum[b * Ssz + tw], pt);
            atomicAdd(&c_cnt[b * Ssz + tw], 1.0f);
        }
    }
}

// ---------------------------------------------------------------------------
// Kernel 2: logits[b,s,t] = a[b,s] + c[b,t] + bias as a rank-2 WMMA outer
// product. One wave per 16x16 tile; V_WMMA_F32_16X16X4_F32 with
//   A (16x4) = [a | 1 | 0 | 0],  B (4x16) = [1 ; c ; 0 ; 0],  C = bias
// => D[m,n] = a[m]*1 + 1*c[n] + bias  (exact in f32).
// Operand construction is branchless (v_cndmask) so EXEC stays all-ones.
// A layout (ISA 7.12.2, 32-bit A 16x4): lanes 0-15 hold K=0/1 in VGPR0/1,
// lanes 16-31 hold K=2/3 (we zero those). C/D layout: VGPR r = row r
// (lanes 0-15) / row r+8 (lanes 16-31), col = lane&15.
// ---------------------------------------------------------------------------
__global__ __launch_bounds__(256) void tile_wmma_kernel(
    const float* __restrict__ a_sum, const float* __restrict__ a_cnt,
    const float* __restrict__ c_sum, const float* __restrict__ c_cnt,
    const float* __restrict__ bias, float* __restrict__ out)
{
    const int lane = threadIdx.x & 31;
    const int wave = threadIdx.x >> 5;
    const int tile = blockIdx.x * 8 + wave;          // 0 .. 16383
    const int b    = tile >> 10;                     // 1024 tiles per batch
    const int st   = tile & 1023;
    const int s0   = (st >> 5) << 4;                 // 32 tile-cols per row
    const int t0   = (st & 31) << 4;

    const int  l    = lane & 15;
    const bool lo   = lane < 16;
    const int  base = b * Ssz;

    // All 32 lanes load (lanes 16-31 redundantly re-read the same 16 values;
    // keeps the loads unpredicated so EXEC is never masked before the WMMA).
    const float asv = a_sum[base + s0 + l];
    const float acv = a_cnt[base + s0 + l];
    const float csv = c_sum[base + t0 + l];
    const float ccv = c_cnt[base + t0 + l];
    const float av  = asv / fmaxf(acv, 1.0f);        // segment mean (clip cnt>=1)
    const float cv  = csv / fmaxf(ccv, 1.0f);
    const float bv  = bias[0];

    v2f Am, Bm;
    Am.x = lo ? av   : 0.0f;   // A[m][0] = a[s0+m]
    Am.y = lo ? 1.0f : 0.0f;   // A[m][1] = 1
    Bm.x = lo ? 1.0f : 0.0f;   // B[0][n] = 1
    Bm.y = lo ? cv   : 0.0f;   // B[1][n] = c[t0+n]

    v8f Cm;
#pragma unroll
    for (int i = 0; i < 8; ++i) Cm[i] = bv;

    v8f D = __builtin_amdgcn_wmma_f32_16x16x4_f32(
        /*neg_a=*/false, Am, /*neg_b=*/false, Bm,
        /*c_mod=*/(short)0, Cm, /*reuse_a=*/false, /*reuse_b=*/false);

    float* __restrict__ op = out + ((size_t)b * Ssz + s0) * Ssz + t0;
    const int rb = lo ? 0 : 8;
#pragma unroll
    for (int r = 0; r < 8; ++r)
        op[(size_t)(r + rb) * Ssz + l] = D[r];       // 64B contiguous per half-wave
}

// ---------------------------------------------------------------------------
// Launch: inputs in setup_inputs() order.
// d_in[0] input_ids (i32, B*L)     d_in[1] attention_mask (i32, B*L)
// d_in[2] source_word_ids (i64)    d_in[3] target_word_ids (i64)
// d_in[4] embed (f32, V*H)         d_in[5] classifier_w (f32, 2H)
// d_in[6] classifier_b (f32, 1)
// d_out: (B,S,S) f32. Workspace: 4 * B*S floats = 128 KB.
// ---------------------------------------------------------------------------
extern "C" void kernel_launch(void* const* d_in, const int* in_sizes, int n_in,
                              void* d_out, int out_size, void* d_ws, size_t ws_size,
                              hipStream_t stream) {
    const int*       ids   = (const int*)d_in[0];
    const int*       mask  = (const int*)d_in[1];
    const long long* swid  = (const long long*)d_in[2];
    const long long* twid  = (const long long*)d_in[3];
    const float*     embed = (const float*)d_in[4];
    const float*     w     = (const float*)d_in[5];
    const float*     bias  = (const float*)d_in[6];
    float*           out   = (float*)d_out;

    float* ws    = (float*)d_ws;
    float* a_sum = ws;                 // B*S = 8192
    float* a_cnt = ws + 1 * Bsz * Ssz;
    float* c_sum = ws + 2 * Bsz * Ssz;
    float* c_cnt = ws + 3 * Bsz * Ssz;
    const int nzero = 4 * Bsz * Ssz;   // 32768 floats

    zero_ws_kernel<<<(nzero + 255) / 256, 256, 0, stream>>>(ws, nzero);

    // B*L = 16384 tokens, 8 waves/block -> 2048 blocks
    token_dot_kernel<<<(Bsz * Lsz) / 8, 256, 0, stream>>>(
        ids, mask, swid, twid, embed, w, a_sum, a_cnt, c_sum, c_cnt);

    // 16 * 32 * 32 = 16384 tiles, 8 waves/block -> 2048 blocks
    tile_wmma_kernel<<<(Bsz * (Ssz / 16) * (Ssz / 16)) / 8, 256, 0, stream>>>(
        a_sum, a_cnt, c_sum, c_cnt, bias, out);
}